// NeighborhoodTokenizer_31894427140137
// MI455X (gfx1250) — compile-verified
//
#include <hip/hip_runtime.h>
#include <hip/hip_bf16.h>

typedef float v4f __attribute__((ext_vector_type(4)));

#define MAXLEN 32   // max_length tokens
#define TOKDIM 64   // token dim
#define SPD    61   // spatial dim
#define TT     64   // timesteps per block tile
#define THREADS 256

// ---- CDNA5 async global->LDS path (ASYNCcnt), guarded by __has_builtin ----
#if defined(__HIP_DEVICE_COMPILE__) && __has_builtin(__builtin_amdgcn_global_load_async_to_lds_b32)
#define HAVE_ASYNC_LDS 1
__device__ __forceinline__ void async_g2l_b32(const void* g, void* l) {
  // prototype (per hipcc diagnostic): (int* gaddr, int* ldsaddr, imm offset, imm cpol)
  __builtin_amdgcn_global_load_async_to_lds_b32((int*)g, (int*)l, /*offset=*/0, /*cpol=*/0);
}
__device__ __forceinline__ void wait_async0() {
#if __has_builtin(__builtin_amdgcn_s_wait_asynccnt)
  __builtin_amdgcn_s_wait_asynccnt(0);
#else
  asm volatile("s_wait_asynccnt 0" ::: "memory");
#endif
}
#endif

__global__ __launch_bounds__(THREADS)
void NeighborhoodTokenizer_kernel(const float* __restrict__ values,   // [num_nodes, n]
                                  const float* __restrict__ tim_emb,  // [n, 2]
                                  const float* __restrict__ spatial,  // [num_nodes, 61]
                                  const float* __restrict__ mu,
                                  const float* __restrict__ sigma,
                                  const int*   __restrict__ nbr_idx,  // [m]
                                  float* __restrict__ out,            // [n, 32, 64]
                                  int n, int m)
{
  __shared__ float s_tmpl[MAXLEN * TOKDIM];  // 8 KB static token template
  __shared__ float s_z[TT][MAXLEN];          // raw gathered sensor values per tile
  __shared__ float s_te[TT][2];              // temporal embeddings per tile
  __shared__ int   s_nbr[MAXLEN];

  const int tid = threadIdx.x;
  const int t0  = blockIdx.x * TT;

  // ---- build static template once per block ----
  if (tid < m) s_nbr[tid] = nbr_idx[tid];
  for (int i = tid; i < MAXLEN * TOKDIM; i += THREADS) s_tmpl[i] = 0.0f;
  __syncthreads();
  for (int i = tid; i < m * SPD; i += THREADS) {
    int j = i / SPD, d = i - j * SPD;
    s_tmpl[j * TOKDIM + d] = spatial[(size_t)s_nbr[j] * SPD + d];
  }
  __syncthreads();

  const float muv = mu[0];
  const float isg = 1.0f / sigma[0];

  // Each thread owns two constant float4 chunks of the 32x64 tile:
  // q0 = tid, q1 = tid + 256; j = q>>4 (token row), c = q&15 (float4 column).
  const int c  = tid & 15;
  const int j0 = tid >> 4;        // 0..15
  const int j1 = j0 + 16;         // 16..31
  v4f a0 = ((const v4f*)s_tmpl)[tid];
  v4f a1 = ((const v4f*)s_tmpl)[tid + THREADS];
  const bool patch = (c == 15);   // chunk covering d = 60..63

  const int ttmax = (n - t0 < TT) ? (n - t0) : TT;

  // ---- stage dynamic data for this tile (async copy to LDS on gfx1250) ----
  for (int i = tid; i < m * TT; i += THREADS) {
    int j  = i >> 6;              // TT == 64
    int tt = i & (TT - 1);        // consecutive lanes -> consecutive t: coalesced
    if (t0 + tt < n) {
      const float* g = values + (size_t)s_nbr[j] * n + (t0 + tt);
#ifdef HAVE_ASYNC_LDS
      async_g2l_b32(g, &s_z[tt][j]);
#else
      s_z[tt][j] = *g;
#endif
    }
  }
  for (int i = tid; i < TT * 2; i += THREADS) {
    int tt = i >> 1, cc = i & 1;
    if (t0 + tt < n) {
      const float* g = tim_emb + (size_t)(t0 + tt) * 2 + cc;
#ifdef HAVE_ASYNC_LDS
      async_g2l_b32(g, &s_te[tt][cc]);
#else
      s_te[tt][cc] = *g;
#endif
    }
  }
#ifdef HAVE_ASYNC_LDS
  wait_async0();
#endif
  __syncthreads();

  // ---- streaming non-temporal stores: 64 timesteps * 8 KB per block ----
  for (int tt = 0; tt < ttmax; ++tt) {
    v4f w0 = a0, w1 = a1;
    if (patch) {
      float te0 = s_te[tt][0], te1 = s_te[tt][1];
      if (j0 < m) { w0.y = (s_z[tt][j0] - muv) * isg; w0.z = te0; w0.w = te1; }
      if (j1 < m) { w1.y = (s_z[tt][j1] - muv) * isg; w1.z = te0; w1.w = te1; }
    }
    size_t base = (size_t)(t0 + tt) * (MAXLEN * TOKDIM);
    __builtin_nontemporal_store(w0, (v4f*)(out + base) + tid);
    __builtin_nontemporal_store(w1, (v4f*)(out + base) + tid + THREADS);
  }
}

extern "C" void kernel_launch(void* const* d_in, const int* in_sizes, int n_in,
                              void* d_out, int out_size, void* d_ws, size_t ws_size,
                              hipStream_t stream) {
  const float* values = (const float*)d_in[0];
  const float* tim    = (const float*)d_in[1];
  const float* sp     = (const float*)d_in[2];
  const float* mu     = (const float*)d_in[3];
  const float* sg     = (const float*)d_in[4];
  const int*   nbr    = (const int*)d_in[5];

  int n = in_sizes[1] / 2;     // tim_emb is [n, 2]
  int m = in_sizes[5];         // self + neighbors (25)
  if (m > MAXLEN) m = MAXLEN;

  int tiles = (n + TT - 1) / TT;
  NeighborhoodTokenizer_kernel<<<tiles, THREADS, 0, stream>>>(
      values, tim, sp, mu, sg, nbr, (float*)d_out, n, m);
}